// PNALayer_28973849378879
// MI455X (gfx1250) — compile-verified
//
#include <hip/hip_runtime.h>
#include <hip/hip_bf16.h>

// ---------------------------------------------------------------------------
// PNA layer:  out = LayerNorm( segment_sum( (x @ W^T + b)[src], dst ) )
// Stage 1: WMMA fp32 GEMM (V_WMMA_F32_16X16X4_F32), Stage 2: edge scatter-add
// with native global_atomic_add_f32 (dev scope), Stage 3: wave-per-row LN.
// ---------------------------------------------------------------------------

typedef __attribute__((ext_vector_type(2))) float v2f;
typedef __attribute__((ext_vector_type(8))) float v8f;

#define D_DIM 64

// ----------------------------- Stage 1: GEMM -------------------------------
// out[m,n] = sum_k x[m,k] * W[n,k] + b[n]
// One wave computes a 16x16 tile of `out`; K = 64 -> 16 chained WMMA 16x16x4.
//
// A-frag (16x4 fp32, ISA 7.12.2): lanes 0-15 hold M=0..15 with K = k0,k0+1;
//                                 lanes 16-31 hold M=0..15 with K = k0+2,k0+3.
// B-frag (4x16 fp32): lanes 0-15 hold N=0..15 rows K=k0,k0+1 (mirrored halves).
//   B[k][n] = W^T[k][n] = W[n][k]  -> contiguous float2 from row n of W.
// D (16x16 f32, 8 VGPRs): vgpr v, lanes 0-15 -> (M=v, N=l); lanes16-31 -> (M=v+8).
__global__ __launch_bounds__(256) void pna_gemm_wmma(
    const float* __restrict__ x, const float* __restrict__ W,
    const float* __restrict__ bias, float* __restrict__ out, int nNodes) {
  const int lane = threadIdx.x & 31;
  const int waveG = (blockIdx.x * 256 + threadIdx.x) >> 5;
  const int nTile = waveG & 3;         // 4 tiles cover D=64 columns
  const int rowTile = waveG >> 2;
  if (rowTile * 16 >= nNodes) return;  // wave-uniform: EXEC stays all-ones

  const int row0 = rowTile * 16;
  const int n0 = nTile * 16;
  const int half = lane >> 4;          // 0: K+{0,1}, 1: K+{2,3}
  const int l = lane & 15;

  const float* xrow = x + (size_t)(row0 + l) * D_DIM;  // A rows, one per lane
  const float* wrow = W + (size_t)(n0 + l) * D_DIM;    // W row n (= B column n)

  v8f acc = {};
#pragma unroll
  for (int k0 = 0; k0 < D_DIM; k0 += 4) {
    const int ka = k0 + half * 2;
    v2f afrag = {xrow[ka], xrow[ka + 1]};
    v2f bfrag = {wrow[ka], wrow[ka + 1]};
    acc = __builtin_amdgcn_wmma_f32_16x16x4_f32(
        /*neg_a=*/false, afrag, /*neg_b=*/false, bfrag,
        /*c_mod=*/(short)0, acc, /*reuse_a=*/false, /*reuse_b=*/false);
  }

  const float bn = bias[n0 + l];
  float* outBase = out + n0 + l;
#pragma unroll
  for (int v = 0; v < 8; ++v) {
    const int m = v + half * 8;
    outBase[(size_t)(row0 + m) * D_DIM] = acc[v] + bn;
  }
}

// --------------------------- Stage 2: scatter ------------------------------
__device__ __forceinline__ void gadd_f32_dev(float* p, float v) {
  // Non-returning fp32 global atomic, device scope (cross-WGP correctness).
  asm volatile("global_atomic_add_f32 %0, %1, off scope:SCOPE_DEV"
               :
               : "v"(p), "v"(v)
               : "memory");
}

__global__ __launch_bounds__(256) void pna_scatter_add(
    const float* __restrict__ msg, const int* __restrict__ src,
    const int* __restrict__ dst, float* __restrict__ agg, int nEdges) {
  const long long gid = (long long)blockIdx.x * 256 + threadIdx.x;
  const long long e = gid >> 6;   // 64 lanes per edge (wave-uniform edge id)
  const int d = (int)(gid & 63);
  if (e >= nEdges) return;
  const int s = src[e];
  const int t = dst[e];
  const float v = msg[(size_t)s * D_DIM + d];
  gadd_f32_dev(agg + (size_t)t * D_DIM + d, v);
}

// -------------------------- Stage 3: layernorm -----------------------------
__global__ __launch_bounds__(256) void pna_layernorm(
    const float* __restrict__ agg, const float* __restrict__ gamma,
    const float* __restrict__ beta, float* __restrict__ out, int nNodes) {
  const int lane = threadIdx.x & 31;
  const int node = (int)(((long long)blockIdx.x * 256 + threadIdx.x) >> 5);
  if (node >= nNodes) return;

  const float2 v = ((const float2*)(agg + (size_t)node * D_DIM))[lane];
  float s = v.x + v.y;
  float ss = v.x * v.x + v.y * v.y;
#pragma unroll
  for (int o = 16; o > 0; o >>= 1) {
    s += __shfl_xor(s, o, 32);
    ss += __shfl_xor(ss, o, 32);
  }
  const float mean = s * (1.0f / D_DIM);
  const float var = ss * (1.0f / D_DIM) - mean * mean;
  const float inv = rsqrtf(var + 1e-5f);

  const float2 g = ((const float2*)gamma)[lane];
  const float2 bt = ((const float2*)beta)[lane];
  float2 o2;
  o2.x = (v.x - mean) * inv * g.x + bt.x;
  o2.y = (v.y - mean) * inv * g.y + bt.y;
  ((float2*)(out + (size_t)node * D_DIM))[lane] = o2;
}

// ------------------------------- launcher ----------------------------------
extern "C" void kernel_launch(void* const* d_in, const int* in_sizes, int n_in,
                              void* d_out, int out_size, void* d_ws, size_t ws_size,
                              hipStream_t stream) {
  const float* x     = (const float*)d_in[0];
  const int*   ei    = (const int*)d_in[1];   // [2, E] flat; jax demotes int64->int32
  const float* W     = (const float*)d_in[2];
  const float* bias  = (const float*)d_in[3];
  const float* gamma = (const float*)d_in[4];
  const float* beta  = (const float*)d_in[5];
  float*       out   = (float*)d_out;

  const int nNodes = in_sizes[0] / D_DIM;   // 100000
  const int nEdges = in_sizes[1] / 2;       // 1200000

  float* t_msg = (float*)d_ws;                       // [N, 64] transformed feats
  float* t_agg = t_msg + (size_t)nNodes * D_DIM;     // [N, 64] accumulator

  hipMemsetAsync(t_agg, 0, (size_t)nNodes * D_DIM * sizeof(float), stream);

  // GEMM: (N/16 row tiles) x (4 col tiles) waves, 8 waves per 256-thread block.
  {
    const long long waves = (long long)((nNodes + 15) / 16) * 4;
    const int blocks = (int)((waves + 7) / 8);
    pna_gemm_wmma<<<blocks, 256, 0, stream>>>(x, W, bias, t_msg, nNodes);
  }

  // Scatter: one lane per (edge, feature).
  {
    const long long threads = (long long)nEdges * D_DIM;
    const int blocks = (int)((threads + 255) / 256);
    pna_scatter_add<<<blocks, 256, 0, stream>>>(t_msg, ei, ei + nEdges, t_agg, nEdges);
  }

  // LayerNorm: one wave32 per node.
  {
    const int blocks = (nNodes + 7) / 8;
    pna_layernorm<<<blocks, 256, 0, stream>>>(t_agg, gamma, beta, out, nNodes);
  }
}